// GauntTensorProduct_LCT_74526272520555
// MI455X (gfx1250) — compile-verified
//
#include <hip/hip_runtime.h>
#include <hip/hip_bf16.h>

typedef __attribute__((ext_vector_type(2))) float v2f;
typedef __attribute__((ext_vector_type(4))) float v4f;
typedef __attribute__((ext_vector_type(8))) float v8f;

#define WNUM 4228
#define O1   128
#define O2   132

// ---------------------------------------------------------------------------
// Kernel 0: precompute G[l][i'][i] (4 x 16 x 16 real) from the complex
// constant tensors. G_l[i',i] = Re( sum_{v,u1,u2:u1+u2<7}
//                                   HC[i',u1,v] * RC[l,u2] * coef[i,u1+u2,v] )
// with coef[i,u,v] = RCuvi2l[i, u, 3+v].
// ---------------------------------------------------------------------------
__global__ void gaunt_gprep(const float* __restrict__ HC,   // (16,7,4,2)
                            const float* __restrict__ RC,   // (4,7,2)
                            const float* __restrict__ R2L,  // (49,7,13,2)
                            float* __restrict__ G)          // (4,16,16)
{
    int t = blockIdx.x * blockDim.x + threadIdx.x;
    if (t >= 1024) return;
    int l  = t >> 8;
    int ip = (t >> 4) & 15;
    int i  = t & 15;
    float acc = 0.f;
    for (int v = 0; v < 4; ++v) {
        for (int u1 = 0; u1 < 7; ++u1) {
            float hr = HC[((ip*7 + u1)*4 + v)*2 + 0];
            float hx = HC[((ip*7 + u1)*4 + v)*2 + 1];
            for (int u2 = 0; u2 < 7 - u1; ++u2) {
                float rr = RC[(l*7 + u2)*2 + 0];
                float rx = RC[(l*7 + u2)*2 + 1];
                float cr = R2L[((i*7 + (u1 + u2))*13 + 3 + v)*2 + 0];
                float cx = R2L[((i*7 + (u1 + u2))*13 + 3 + v)*2 + 1];
                float pr = hr*rr - hx*rx;    // Re(HC*RC)
                float px = hr*rx + hx*rr;    // Im(HC*RC)
                acc += pr*cr - px*cx;        // Re((HC*RC)*coef)
            }
        }
    }
    G[t] = acc;
}

__device__ __forceinline__ int l_of(int i) {
    return (i >= 9) ? 3 : ((i >= 4) ? 2 : ((i >= 1) ? 1 : 0));
}

// ---------------------------------------------------------------------------
// Main kernel: one wave32 per batch element.
//   g  = rot(feature) * weight1            (per-lane VALU, lane = channel c)
//   A  = sum_l w2p[l] * G_l                (16x16, in LDS)
//   f2 = g @ A                             (WMMA f32 16x16x4, M=32,K=16,N=16)
//   h  = inv-rot(f2)                       (per-lane VALU)
//   O  = wc^T @ B'                         (WMMA f32 16x16x4, M=32,K=128,N=16)
//        B'[(l,c),i] = h[c,i]*[l==l(i)]  (block-diagonal stack in LDS)
// ---------------------------------------------------------------------------
__global__ __launch_bounds__(32)
void gaunt_main(const float* __restrict__ feature,
                const float* __restrict__ weight,
                const float* __restrict__ wD0, const float* __restrict__ wD1,
                const float* __restrict__ wD2, const float* __restrict__ wD3,
                const unsigned char* __restrict__ mask_edge,
                const unsigned char* __restrict__ mask_sc,
                const float* __restrict__ G,
                float* __restrict__ out)
{
    __shared__ float sD[84];         // wD blocks: off {0,1,10,35}, d {1,3,5,7}
    __shared__ float sA[256];        // 16x16 mixing matrix (row-major i' x i)
    __shared__ float sGst[32*17];    // g, then f2 (row stride 17: bank-safe)
    __shared__ float sB[128*17];     // B' (128x16); also feature staging (512f)

    const int lane = threadIdx.x;
    const int b    = blockIdx.x;
    const int hi   = lane >> 4;      // which 16-lane half
    const int ln   = lane & 15;

    const float* wgt = weight + (size_t)b * WNUM;

    // ---- stage wD into LDS ----
    for (int t = lane; t < 84; t += 32) {
        float v;
        if      (t < 1)  v = wD0[(size_t)b*1  + t];
        else if (t < 10) v = wD1[(size_t)b*9  + (t - 1)];
        else if (t < 35) v = wD2[(size_t)b*25 + (t - 10)];
        else             v = wD3[(size_t)b*49 + (t - 35)];
        sD[t] = v;
    }

    // ---- stage feature row (512 floats) into sB, coalesced b128 ----
    {
        const v4f* fsrc = (const v4f*)(feature + (size_t)b * 512);
        v4f* fdst = (v4f*)sB;
        #pragma unroll
        for (int t = 0; t < 4; ++t) fdst[lane + 32*t] = fsrc[lane + 32*t];
    }

    // ---- A = sum_l w2p[l] * G_l ----
    {
        bool me = mask_edge[b] != 0;
        bool ms = mask_sc[b]  != 0;
        float w2p[4];
        #pragma unroll
        for (int l = 0; l < 4; ++l) {
            float sh = sqrtf((2.f*l + 1.f) * (1.f / (4.f * 3.14159265358979323846f)));
            float pref = me ? sh : ((ms && l == 0) ? sh : 0.f);
            w2p[l] = wgt[O1 + l] * pref;
        }
        #pragma unroll
        for (int j = 0; j < 8; ++j) {
            int e = lane * 8 + j;
            float acc = 0.f;
            #pragma unroll
            for (int l = 0; l < 4; ++l) acc += w2p[l] * G[l*256 + e];
            sA[e] = acc;
        }
    }

    __syncthreads();

    // ---- per-lane (lane = channel c): rotate feature, apply weight1 ----
    {
        const int c = lane;
        float ft[16], f[16];
        ft[0] = sB[c];
        #pragma unroll
        for (int j = 0; j < 3; ++j) ft[1+j] = sB[32  + 3*c + j];
        #pragma unroll
        for (int j = 0; j < 5; ++j) ft[4+j] = sB[128 + 5*c + j];
        #pragma unroll
        for (int j = 0; j < 7; ++j) ft[9+j] = sB[288 + 7*c + j];

        // f[l^2+i] = sum_j D_l[j][i] * ft[l^2+j]
        f[0] = sD[0] * ft[0];
        #pragma unroll
        for (int i = 0; i < 3; ++i) {
            float a = 0.f;
            #pragma unroll
            for (int j = 0; j < 3; ++j) a += sD[1 + j*3 + i] * ft[1+j];
            f[1+i] = a;
        }
        #pragma unroll
        for (int i = 0; i < 5; ++i) {
            float a = 0.f;
            #pragma unroll
            for (int j = 0; j < 5; ++j) a += sD[10 + j*5 + i] * ft[4+j];
            f[4+i] = a;
        }
        #pragma unroll
        for (int i = 0; i < 7; ++i) {
            float a = 0.f;
            #pragma unroll
            for (int j = 0; j < 7; ++j) a += sD[35 + j*7 + i] * ft[9+j];
            f[9+i] = a;
        }

        float w1v[4];
        #pragma unroll
        for (int l = 0; l < 4; ++l) w1v[l] = wgt[c*4 + l];
        #pragma unroll
        for (int i = 0; i < 16; ++i)
            sGst[c*17 + i] = f[i] * w1v[l_of(i)];
    }

    __syncthreads();

    // ---- f2 = g @ A  (M=32 -> 2 tiles, K=16 -> 4 steps of 16x16x4 f32) ----
    v8f f2t0 = {}, f2t1 = {};
    #pragma unroll
    for (int k = 0; k < 4; ++k) {
        int kr = 4*k + 2*hi;
        v2f bb;
        bb.x = sA[kr*16 + ln];
        bb.y = sA[(kr+1)*16 + ln];
        v2f a0, a1;
        a0.x = sGst[ln*17 + kr];        a0.y = sGst[ln*17 + kr + 1];
        a1.x = sGst[(16+ln)*17 + kr];   a1.y = sGst[(16+ln)*17 + kr + 1];
        f2t0 = __builtin_amdgcn_wmma_f32_16x16x4_f32(false, a0, false, bb, (short)0, f2t0, false, false);
        f2t1 = __builtin_amdgcn_wmma_f32_16x16x4_f32(false, a1, false, bb, (short)0, f2t1, false, false);
    }

    __syncthreads();

    // write f2 back into sGst (C/D layout: vgpr r -> row r + 8*hi, col ln)
    #pragma unroll
    for (int r = 0; r < 8; ++r) {
        sGst[(r + 8*hi)*17 + ln]      = f2t0[r];
        sGst[(16 + r + 8*hi)*17 + ln] = f2t1[r];
    }

    __syncthreads();

    // ---- per-lane: inverse rotation h[c,i] = sum_j D_l[i][j] f2[c, l^2+j];
    //      scatter into block-diagonal B' (128 x 16) ----
    {
        const int c = lane;
        float ff[16], h[16];
        #pragma unroll
        for (int i = 0; i < 16; ++i) ff[i] = sGst[c*17 + i];

        h[0] = sD[0] * ff[0];
        #pragma unroll
        for (int i = 0; i < 3; ++i) {
            float a = 0.f;
            #pragma unroll
            for (int j = 0; j < 3; ++j) a += sD[1 + i*3 + j] * ff[1+j];
            h[1+i] = a;
        }
        #pragma unroll
        for (int i = 0; i < 5; ++i) {
            float a = 0.f;
            #pragma unroll
            for (int j = 0; j < 5; ++j) a += sD[10 + i*5 + j] * ff[4+j];
            h[4+i] = a;
        }
        #pragma unroll
        for (int i = 0; i < 7; ++i) {
            float a = 0.f;
            #pragma unroll
            for (int j = 0; j < 7; ++j) a += sD[35 + i*7 + j] * ff[9+j];
            h[9+i] = a;
        }

        #pragma unroll
        for (int l = 0; l < 4; ++l) {
            int row = l*32 + c;
            #pragma unroll
            for (int i = 0; i < 16; ++i)
                sB[row*17 + i] = (l_of(i) == l) ? h[i] : 0.f;
        }
    }

    __syncthreads();

    // ---- O = wc^T @ B'  (M=32 -> 2 tiles, K=128 -> 32 steps, N=16) ----
    // A-operand streamed directly from global: A''[a][K] = wgt[O2 + K*32 + a]
    v8f o0 = {}, o1 = {};
    const float* wcb = wgt + O2;
    #pragma unroll 4
    for (int k = 0; k < 32; ++k) {
        int k0 = 4*k + 2*hi;
        v2f bb;
        bb.x = sB[k0*17 + ln];
        bb.y = sB[(k0+1)*17 + ln];
        v2f a0, a1;
        a0.x = wcb[k0*32 + ln];        a0.y = wcb[(k0+1)*32 + ln];
        a1.x = wcb[k0*32 + 16 + ln];   a1.y = wcb[(k0+1)*32 + 16 + ln];
        o0 = __builtin_amdgcn_wmma_f32_16x16x4_f32(false, a0, false, bb, (short)0, o0, false, false);
        o1 = __builtin_amdgcn_wmma_f32_16x16x4_f32(false, a1, false, bb, (short)0, o1, false, false);
    }

    // ---- scatter output: out[b, 32*l^2 + a*d + iloc], col ln fixed per lane ----
    {
        int l    = l_of(ln);
        int dd   = 2*l + 1;
        int iloc = ln - l*l;
        float* op = out + (size_t)b*512 + 32*l*l;
        #pragma unroll
        for (int r = 0; r < 8; ++r) {
            op[(r + 8*hi)*dd + iloc]      = o0[r];
            op[(16 + r + 8*hi)*dd + iloc] = o1[r];
        }
    }
}

extern "C" void kernel_launch(void* const* d_in, const int* in_sizes, int n_in,
                              void* d_out, int out_size, void* d_ws, size_t ws_size,
                              hipStream_t stream) {
    const float* feature = (const float*)d_in[0];
    const float* weight  = (const float*)d_in[1];
    const float* wD0 = (const float*)d_in[2];
    const float* wD1 = (const float*)d_in[3];
    const float* wD2 = (const float*)d_in[4];
    const float* wD3 = (const float*)d_in[5];
    const unsigned char* mask_edge = (const unsigned char*)d_in[6];
    const unsigned char* mask_sc  = (const unsigned char*)d_in[7];
    const float* HC  = (const float*)d_in[8];
    const float* RC  = (const float*)d_in[9];
    const float* R2L = (const float*)d_in[10];
    float* outp = (float*)d_out;
    float* G    = (float*)d_ws;            // 4*16*16 floats = 4 KB scratch

    int B = in_sizes[0] / 512;             // feature is (B, 512)

    gaunt_gprep<<<4, 256, 0, stream>>>(HC, RC, R2L, G);
    gaunt_main<<<B, 32, 0, stream>>>(feature, weight, wD0, wD1, wD2, wD3,
                                     mask_edge, mask_sc, G, outp);
}